// GraphAttentionV2Layer_40140764348523
// MI455X (gfx1250) — compile-verified
//
#include <hip/hip_runtime.h>
#include <math.h>

typedef float v2f __attribute__((ext_vector_type(2)));
typedef float v8f __attribute__((ext_vector_type(8)));

#define N_NODES 1024
#define IN_F    256
#define OUT_F   256
#define HEADS   4
#define HID     64
#define NEG_SLOPE 0.2f

// ---------------------------------------------------------------------------
// Kernel 1: G = X @ W   (1024x256 @ 256x256, f32), one 16x16 C-tile per wave
// using V_WMMA_F32_16X16X4_F32.
// A layout (16x4 f32): lane l, vgpr p -> A[M = l%16][K = k0 + 2*(l/16) + p]
// B layout (4x16 f32): lane l, vgpr p -> B[K = k0 + 2*(l/16) + p][N = l%16]
// C layout (16x16 f32): lane l, vgpr p -> C[M = p + 8*(l/16)][N = l%16]
// ---------------------------------------------------------------------------
__global__ void gemm_xw(const float* __restrict__ X, const float* __restrict__ W,
                        float* __restrict__ G) {
    const int lane  = threadIdx.x;          // 0..31 (one wave per block)
    const int tileN = blockIdx.x * 16;
    const int tileM = blockIdx.y * 16;
    const int lid   = lane & 15;
    const int kh    = (lane >> 4) * 2;      // 0 or 2
    const int m     = tileM + lid;
    const int n     = tileN + lid;

    v8f acc = {};
    for (int k0 = 0; k0 < IN_F; k0 += 4) {
        const int k = k0 + kh;
        v2f a, b;
        a.x = X[m * IN_F + k];
        a.y = X[m * IN_F + k + 1];
        b.x = W[k * OUT_F + n];
        b.y = W[(k + 1) * OUT_F + n];
        acc = __builtin_amdgcn_wmma_f32_16x16x4_f32(false, a, false, b,
                                                    (short)0, acc, false, false);
    }
    const int row_half = (lane >> 4) * 8;
#pragma unroll
    for (int p = 0; p < 8; ++p) {
        const int mm = tileM + p + row_half;
        G[mm * OUT_F + n] = acc[p];
    }
}

// ---------------------------------------------------------------------------
// Kernel 2: masked exp-attention scores + row denominators.
// One block per (i, h).  P[h][i][j] = adj[j][i] ? exp(e[i,j,h]) : 0
// denom[h][i] = sum_j P[h][i][j]
// ---------------------------------------------------------------------------
__global__ void attn_scores(const float* __restrict__ gl, const float* __restrict__ gr,
                            const float* __restrict__ adj, const float* __restrict__ aw,
                            float* __restrict__ P, float* __restrict__ denom) {
    const int i = blockIdx.x;
    const int h = blockIdx.y;
    const int t = threadIdx.x;              // 256 threads

    __shared__ float s_gri[HID];
    __shared__ float s_aw[HID];
    __shared__ float s_red[256];

    if (t < HID) {
        s_gri[t] = gr[i * OUT_F + h * HID + t];
        s_aw[t]  = aw[t];
    }
    __syncthreads();

    float part = 0.0f;
    float* Ph = P + (size_t)h * N_NODES * N_NODES + (size_t)i * N_NODES;
    for (int j = t; j < N_NODES; j += 256) {
        float p = 0.0f;
        if (adj[(size_t)j * N_NODES + i] != 0.0f) {
            const float* glj = gl + j * OUT_F + h * HID;
            float e = 0.0f;
#pragma unroll
            for (int f = 0; f < HID; ++f) {
                const float s  = s_gri[f] + glj[f];
                const float lr = fmaxf(s, NEG_SLOPE * s);   // leaky_relu(s, 0.2)
                e = fmaf(lr, s_aw[f], e);
            }
            p = expf(e);
        }
        Ph[j] = p;
        part += p;
    }

    s_red[t] = part;
    __syncthreads();
    for (int s = 128; s > 0; s >>= 1) {
        if (t < s) s_red[t] += s_red[t + s];
        __syncthreads();
    }
    if (t == 0) denom[h * N_NODES + i] = s_red[0];
}

// ---------------------------------------------------------------------------
// Kernel 3: attn_res = elu( (P/denom) @ g_r ) per head, WMMA f32 16x16x4.
// One 16(i) x 16(f) C-tile per wave, K = 1024 neighbor loop.
// ---------------------------------------------------------------------------
__global__ void aggregate(const float* __restrict__ P, const float* __restrict__ denom,
                          const float* __restrict__ gr, float* __restrict__ out,
                          const int* __restrict__ use_elu) {
    const int lane  = threadIdx.x;          // one wave per block
    const int tileF = blockIdx.x * 16;      // 0..48
    const int tileM = blockIdx.y * 16;      // i tile
    const int h     = blockIdx.z;
    const int lid   = lane & 15;
    const int kh    = (lane >> 4) * 2;
    const int m     = tileM + lid;

    const float recip = 1.0f / denom[h * N_NODES + m];
    const float* Ph = P + (size_t)h * N_NODES * N_NODES;

    v8f acc = {};
    for (int k0 = 0; k0 < N_NODES; k0 += 4) {
        const int k = k0 + kh;
        v2f a, b;
        a.x = Ph[(size_t)m * N_NODES + k]     * recip;
        a.y = Ph[(size_t)m * N_NODES + k + 1] * recip;
        b.x = gr[k * OUT_F + h * HID + tileF + lid];
        b.y = gr[(k + 1) * OUT_F + h * HID + tileF + lid];
        acc = __builtin_amdgcn_wmma_f32_16x16x4_f32(false, a, false, b,
                                                    (short)0, acc, false, false);
    }

    const int elu = use_elu[0];
    const int row_half = (lane >> 4) * 8;
#pragma unroll
    for (int p = 0; p < 8; ++p) {
        const int i = tileM + p + row_half;
        float x = acc[p];
        if (elu) x = (x > 0.0f) ? x : expm1f(x);
        out[i * OUT_F + h * HID + tileF + lid] = x;
    }
}

// ---------------------------------------------------------------------------
// Launcher. Inputs: h, nei_embedding, adj_mat, w_l, w_r, attn_w, use_elu
// ws layout (floats): g_l[1024*256] | g_r[1024*256] | P[4*1024*1024] | denom[4096]
// ---------------------------------------------------------------------------
extern "C" void kernel_launch(void* const* d_in, const int* in_sizes, int n_in,
                              void* d_out, int out_size, void* d_ws, size_t ws_size,
                              hipStream_t stream) {
    const float* h_in  = (const float*)d_in[0];
    const float* nei   = (const float*)d_in[1];
    const float* adj   = (const float*)d_in[2];
    const float* w_l   = (const float*)d_in[3];
    const float* w_r   = (const float*)d_in[4];
    const float* aw    = (const float*)d_in[5];
    const int*   elu   = (const int*)d_in[6];
    float* out = (float*)d_out;

    float* ws    = (float*)d_ws;
    float* g_l   = ws;                                   // 262144
    float* g_r   = ws + (size_t)N_NODES * OUT_F;         // 262144
    float* P     = g_r + (size_t)N_NODES * OUT_F;        // 4*1024*1024
    float* denom = P + (size_t)HEADS * N_NODES * N_NODES; // 4096

    // Stage 1: projections (WMMA f32)
    dim3 ggrid(OUT_F / 16, N_NODES / 16);
    gemm_xw<<<ggrid, 32, 0, stream>>>(h_in, w_l, g_l);
    gemm_xw<<<ggrid, 32, 0, stream>>>(nei,  w_r, g_r);

    // Stage 2: masked exp scores + denominators (VALU)
    attn_scores<<<dim3(N_NODES, HEADS), 256, 0, stream>>>(g_l, g_r, adj, aw, P, denom);

    // Stage 3: softmax-normalized aggregation + ELU (WMMA f32)
    aggregate<<<dim3(HID / 16, N_NODES / 16, HEADS), 32, 0, stream>>>(P, denom, g_r, out, elu);
}